// EMDLoss_13778255085629
// MI455X (gfx1250) — compile-verified
//
#include <hip/hip_runtime.h>

// EMD loss, algebraically collapsed (top_k with k=N => assignment == ones):
//   out = (1/B) * sum_b [ ||pred_b||_F^2 + ||targ_b||_F^2 - (2/N) * <sum_n pred_b, sum_m targ_b> ]
// Single streaming pass over ~1.5 MB; reductions on the WMMA pipe + ds_swizzle.

typedef float v2f __attribute__((ext_vector_type(2)));
typedef float v8f __attribute__((ext_vector_type(8)));

#define NBATCH 64
#define NPTS   1024
#define INV_N  (1.0f / 1024.0f)

// xor-swap-add across the wave via ds_swizzle_b32 (group-of-32 mode):
// offset = {0, xor_mask[14:10], or_mask[9:5]=0, and_mask[4:0]=0x1F}
template <int XOR_MASK>
__device__ __forceinline__ float swz_xor_add(float v) {
    const int pat = (XOR_MASK << 10) | 0x1F;
    int o = __builtin_amdgcn_ds_swizzle(__float_as_int(v), pat);
    return v + __int_as_float(o);
}

// Sum a per-lane scalar across the wave using V_WMMA_F32_16X16X4_F32.
// A = ones(16x4), B slots hold {v, 0} per lane. D[m,n] = colsum(B)[n], so
// the sum over D row 0 == sum over ALL 64 B slots == sum over 32 lanes of v
// (robust to the exact B lane mapping). Row 0 of D is d[0] lanes 0-15; lanes
// 16-31 hold row 8 which is identical, so 4 xor-swizzle adds finish it and
// the total lands in every lane.
__device__ __forceinline__ float wave_sum_wmma(float v) {
    v2f a = {1.0f, 1.0f};     // A = all ones
    v2f b = {v, 0.0f};        // lane partials into B operand
    v8f c = {};
    v8f d = __builtin_amdgcn_wmma_f32_16x16x4_f32(
        /*neg_a=*/false, a, /*neg_b=*/false, b,
        /*c_mod=*/(short)0, c, /*reuse_a=*/false, /*reuse_b=*/false);
    float s = d[0];
    s = swz_xor_add<8>(s);
    s = swz_xor_add<4>(s);
    s = swz_xor_add<2>(s);
    s = swz_xor_add<1>(s);
    return s;
}

// One block per batch. 256 threads x 4 points/thread = 1024 points.
// Each point is a float3; thread t loads 3 aligned float4 = 12 floats per tensor.
__global__ void emd_batch_kernel(const float* __restrict__ pred,
                                 const float* __restrict__ targ,
                                 float* __restrict__ ws) {
    const int b = blockIdx.x;
    const int t = threadIdx.x;

    const float4* p4 = reinterpret_cast<const float4*>(pred) + (size_t)b * 768 + 3 * t;
    const float4* q4 = reinterpret_cast<const float4*>(targ) + (size_t)b * 768 + 3 * t;

    float4 a0 = p4[0], a1 = p4[1], a2 = p4[2];
    float4 c0 = q4[0], c1 = q4[1], c2 = q4[2];

    // sum of squares over this thread's 12+12 floats (p2 and t2 terms combine additively)
    float sq =
        a0.x*a0.x + a0.y*a0.y + a0.z*a0.z + a0.w*a0.w +
        a1.x*a1.x + a1.y*a1.y + a1.z*a1.z + a1.w*a1.w +
        a2.x*a2.x + a2.y*a2.y + a2.z*a2.z + a2.w*a2.w +
        c0.x*c0.x + c0.y*c0.y + c0.z*c0.z + c0.w*c0.w +
        c1.x*c1.x + c1.y*c1.y + c1.z*c1.z + c1.w*c1.w +
        c2.x*c2.x + c2.y*c2.y + c2.z*c2.z + c2.w*c2.w;

    // per-component sums; within 12 consecutive floats, x at 0,3,6,9; y at 1,4,7,10; z at 2,5,8,11
    float comp[6];
    comp[0] = a0.x + a0.w + a1.z + a2.y;   // Sp.x
    comp[1] = a0.y + a1.x + a1.w + a2.z;   // Sp.y
    comp[2] = a0.z + a1.y + a2.x + a2.w;   // Sp.z
    comp[3] = c0.x + c0.w + c1.z + c2.y;   // St.x
    comp[4] = c0.y + c1.x + c1.w + c2.z;   // St.y
    comp[5] = c0.z + c1.y + c2.x + c2.w;   // St.z

    // wave-level reductions: WMMA pipe for the Frobenius term, ds_swizzle for components
    float wsq = wave_sum_wmma(sq);
#pragma unroll
    for (int q = 0; q < 6; ++q) {
        float v = comp[q];
        v = swz_xor_add<16>(v);
        v = swz_xor_add<8>(v);
        v = swz_xor_add<4>(v);
        v = swz_xor_add<2>(v);
        v = swz_xor_add<1>(v);
        comp[q] = v;
    }

    __shared__ float red[8][8];  // [wave][0]=sq, [1..6]=comp
    const int wave = t >> 5;
    const int lane = t & 31;
    if (lane == 0) {
        red[wave][0] = wsq;
#pragma unroll
        for (int q = 0; q < 6; ++q) red[wave][1 + q] = comp[q];
    }
    __syncthreads();

    if (t == 0) {
        float acc[7] = {0.f, 0.f, 0.f, 0.f, 0.f, 0.f, 0.f};
#pragma unroll
        for (int w = 0; w < 8; ++w)
#pragma unroll
            for (int q = 0; q < 7; ++q) acc[q] += red[w][q];
        const float cross = acc[1]*acc[4] + acc[2]*acc[5] + acc[3]*acc[6];
        ws[b] = acc[0] - 2.0f * INV_N * cross;   // emd[b]
    }
}

// One wave folds the 64 per-batch values: both B-operand VGPRs carry data,
// so one WMMA sums all 64 slots into 16 column sums, 4 swizzle-adds finish.
__global__ void emd_final_kernel(const float* __restrict__ ws,
                                 float* __restrict__ out) {
    const int lane = threadIdx.x;  // 0..31
    const float e0 = ws[2 * lane];
    const float e1 = ws[2 * lane + 1];
    v2f a = {1.0f, 1.0f};
    v2f b = {e0, e1};
    v8f c = {};
    v8f d = __builtin_amdgcn_wmma_f32_16x16x4_f32(
        false, a, false, b, (short)0, c, false, false);
    float s = d[0];
    s = swz_xor_add<8>(s);
    s = swz_xor_add<4>(s);
    s = swz_xor_add<2>(s);
    s = swz_xor_add<1>(s);
    if (lane == 0) out[0] = s * (1.0f / (float)NBATCH);
}

extern "C" void kernel_launch(void* const* d_in, const int* in_sizes, int n_in,
                              void* d_out, int out_size, void* d_ws, size_t ws_size,
                              hipStream_t stream) {
    const float* pred = (const float*)d_in[0];
    const float* targ = (const float*)d_in[1];
    float* ws  = (float*)d_ws;    // 64 floats of scratch (per-batch emd)
    float* out = (float*)d_out;   // scalar f32

    emd_batch_kernel<<<NBATCH, 256, 0, stream>>>(pred, targ, ws);
    emd_final_kernel<<<1, 32, 0, stream>>>(ws, out);
}